// LinAngularAttention_27522150432914
// MI455X (gfx1250) — compile-verified
//
#include <hip/hip_runtime.h>
#include <hip/hip_bf16.h>
#include <math.h>

typedef float v2f __attribute__((ext_vector_type(2)));
typedef float v8f __attribute__((ext_vector_type(8)));

#define NB 8
#define CC 256
#define HH 8
#define DD 32
#define LL 4096
#define TC 768  // 3*C

static __device__ __forceinline__ v8f wmma_f32(v2f a, v2f b, v8f c) {
    // D = A(16x4 f32) * B(4x16 f32) + C(16x16 f32)
    return __builtin_amdgcn_wmma_f32_16x16x4_f32(false, a, false, b, (short)0, c, false, false);
}

// -------------------- Kernel 1: qkv = x^T @ w_qkv^T + b_qkv --------------------
// x: [N, C, L]  (A[l][c] = x[n][c][l], column-major in M -> coalesced frag loads)
// w_qkv: [768, 256] row-major; staged in LDS as float2 K-pairs: Wt2[p][j] = {W[2p][j], W[2p+1][j]}
// so each B fragment is ONE ds_load_b64 into an even VGPR pair (no mov shuffles).
// qkv: [N, L, 768]
__global__ __launch_bounds__(256) void qkv_gemm_kernel(
    const float* __restrict__ x, const float* __restrict__ wq,
    const float* __restrict__ bq, float* __restrict__ qkv)
{
    __shared__ float2 Wt2[16 * 66];  // pair rows p=0..15, stride 66 (bank rotation 4/dword-pair)
    const int mt = blockIdx.x;       // 256 tiles of 128 rows
    const int nt = blockIdx.y;       // 12 tiles of 64 cols
    const int row0 = mt * 128;       // global row = n*L + l
    const int n = row0 >> 12;
    const int l0 = row0 & (LL - 1);
    const int j0 = nt * 64;
    const int tid = threadIdx.x;
    const int lane = tid & 31, wave = tid >> 5;
    const int m0 = wave * 16;
    const int laneM = lane & 15;
    const int laneH = lane >> 4;         // 0 or 1 (K-pair select)
    const int laneK = laneH << 1;        // 0 or 2

    v8f acc[4] = {};
    const float* xA = x + (size_t)n * CC * LL + l0 + m0 + laneM;

    for (int kc = 0; kc < CC; kc += 32) {
        // stage: lanes sweep p fastest -> 128B contiguous global reads per j row
        for (int i = tid; i < 16 * 64; i += 256) {
            const int p = i & 15, j = i >> 4;
            Wt2[p * 66 + j] = *(const float2*)&wq[(size_t)(j0 + j) * CC + kc + 2 * p];
        }
        __syncthreads();
#pragma unroll
        for (int kk = 0; kk < 32; kk += 4) {
            const int k = kc + kk + laneK;
            v2f a;
            a.x = xA[(size_t)k * LL];
            a.y = xA[(size_t)(k + 1) * LL];
            const int pp = (kk >> 1) + laneH;  // K-pair index
#pragma unroll
            for (int nb = 0; nb < 4; ++nb) {
                const v2f b = *(const v2f*)&Wt2[pp * 66 + nb * 16 + laneM];
                acc[nb] = wmma_f32(a, b, acc[nb]);
            }
        }
        __syncthreads();
    }
    // Direct coalesced store: D tile lanes 0-15 -> j contiguous.
    const int mbase = m0 + (laneH << 3);
#pragma unroll
    for (int nb = 0; nb < 4; ++nb) {
        const int j = j0 + nb * 16 + laneM;
        const float bias = bq[j];
        float* outp = qkv + (size_t)(row0 + mbase) * TC + j;
#pragma unroll
        for (int r = 0; r < 8; ++r) outp[(size_t)r * TC] = acc[nb][r] + bias;
    }
}

// -------------------- Kernel 2: L2-normalize q and k rows in place --------------------
__global__ __launch_bounds__(256) void norm_qk_kernel(float* __restrict__ qkv)
{
    size_t t = (size_t)blockIdx.x * 256 + threadIdx.x;  // 2*N*L*H rows
    const int h = (int)(t & 7);
    size_t r = t >> 3;
    const int l = (int)(r & (LL - 1));
    r >>= 12;
    const int n = (int)(r & 7);
    const int s = (int)(r >> 3);  // 0 = q, 1 = k
    float* p = qkv + ((size_t)(n * LL + l)) * TC + s * CC + h * DD;
    float4 v[8];
    float ss = 0.f;
#pragma unroll
    for (int i = 0; i < 8; ++i) {
        v[i] = ((const float4*)p)[i];
        ss += v[i].x * v[i].x + v[i].y * v[i].y + v[i].z * v[i].z + v[i].w * v[i].w;
    }
    const float inv = 1.0f / sqrtf(ss);
#pragma unroll
    for (int i = 0; i < 8; ++i) {
        v[i].x *= inv; v[i].y *= inv; v[i].z *= inv; v[i].w *= inv;
        ((float4*)p)[i] = v[i];
    }
}

// -------------------- Kernel 3: attn[n,h] = k^T @ v  (32x32, K=L=4096) --------------------
// One block per (n,h); 8 waves K-split over L; WMMA f32; LDS tree-reduce.
__global__ __launch_bounds__(256) void attn_kernel(
    const float* __restrict__ qkv, float* __restrict__ attn)
{
    __shared__ float red[8 * 1024];
    const int nh = blockIdx.x;
    const int n = nh >> 3, h = nh & 7;
    const int tid = threadIdx.x, lane = tid & 31, wave = tid >> 5;
    const int laneM = lane & 15;
    const int laneK = (lane >> 4) << 1;

    v8f acc[2][2] = {};
    const float* base = qkv + (size_t)n * LL * TC + h * DD;

    const int lend = wave * 512 + 512;
    for (int l = wave * 512; l < lend; l += 4) {
        const size_t r0 = (size_t)(l + laneK) * TC;
        v2f aK[2], bV[2];
#pragma unroll
        for (int md = 0; md < 2; ++md) {
            aK[md].x = base[r0 + CC + md * 16 + laneM];        // k, row l+laneK
            aK[md].y = base[r0 + TC + CC + md * 16 + laneM];   // k, row l+laneK+1
        }
#pragma unroll
        for (int me = 0; me < 2; ++me) {
            bV[me].x = base[r0 + 2 * CC + me * 16 + laneM];    // v
            bV[me].y = base[r0 + TC + 2 * CC + me * 16 + laneM];
        }
#pragma unroll
        for (int md = 0; md < 2; ++md)
#pragma unroll
            for (int me = 0; me < 2; ++me)
                acc[md][me] = wmma_f32(aK[md], bV[me], acc[md][me]);
    }
    // partials -> LDS:  d = md*16 + r + 8*(lane>=16), e = me*16 + laneM
    const int dadd = (lane >> 4) << 3;
#pragma unroll
    for (int md = 0; md < 2; ++md)
#pragma unroll
        for (int me = 0; me < 2; ++me)
#pragma unroll
            for (int r = 0; r < 8; ++r)
                red[wave * 1024 + (md * 16 + r + dadd) * 32 + me * 16 + laneM] = acc[md][me][r];
    __syncthreads();
    for (int i = tid; i < 1024; i += 256) {
        float s = 0.f;
#pragma unroll
        for (int w = 0; w < 8; ++w) s += red[w * 1024 + i];
        attn[(size_t)nh * 1024 + i] = s;
    }
}

// -------------------- Kernel 4: fused epilogue --------------------
// out_pre[n,l,h*32+e] = l2norm(0.5*v + (1/pi) * q@attn) + dconv9(v)
__global__ __launch_bounds__(256) void epilogue_kernel(
    const float* __restrict__ qkv, const float* __restrict__ attn,
    const float* __restrict__ wdc, float* __restrict__ out_pre)
{
    __shared__ float v_s[72 * 32];   // rows l0-4 .. l0+67
    __shared__ float q_s[64 * 32];
    __shared__ float a_s[32 * 32];
    const int b = blockIdx.x;        // n*512 + h*64 + lc
    const int lc = b & 63;
    const int h = (b >> 6) & 7;
    const int n = b >> 9;
    const int l0 = lc * 64;
    const int tid = threadIdx.x;
    const float* qbase = qkv + (size_t)n * LL * TC + h * DD;

    for (int i = tid; i < 72 * 32; i += 256) {
        const int rr = i >> 5, e = i & 31;
        const int l = l0 - 4 + rr;
        v_s[i] = (l >= 0 && l < LL) ? qbase[(size_t)l * TC + 2 * CC + e] : 0.0f;
    }
    for (int i = tid; i < 64 * 32; i += 256) {
        const int rr = i >> 5, d = i & 31;
        q_s[i] = qbase[(size_t)(l0 + rr) * TC + d];
    }
    for (int i = tid; i < 1024; i += 256)
        a_s[i] = attn[((size_t)(n * 8 + h)) * 1024 + i];
    __syncthreads();

    float wd[9];
#pragma unroll
    for (int r = 0; r < 9; ++r) wd[r] = wdc[h * 9 + r];

    const int lane = tid & 31, wave = tid >> 5;
    const float inv_pi = 0.31830988618379067154f;
    for (int ll = wave * 8; ll < wave * 8 + 8; ++ll) {
        float acc = 0.f;
#pragma unroll
        for (int d = 0; d < 32; ++d) acc += q_s[ll * 32 + d] * a_s[d * 32 + lane];
        const float val = 0.5f * v_s[(ll + 4) * 32 + lane] + inv_pi * acc;
        float ss = val * val;
#pragma unroll
        for (int off = 16; off; off >>= 1) ss += __shfl_xor(ss, off, 32);
        const float res = val / sqrtf(ss);
        float dc = 0.f;
#pragma unroll
        for (int r = 0; r < 9; ++r) dc += wd[r] * v_s[(ll + r) * 32 + lane];
        out_pre[(size_t)(n * LL + l0 + ll) * CC + h * DD + lane] = res + dc;
    }
}

// -------------------- Kernel 5: out[n,j,l] = out_pre @ w_proj^T + b_proj --------------------
// A row-major -> LDS float2 K-pair tiles (single ds_load_b64 per fragment);
// output transposed -> LDS transpose store, coalesced along l.
__global__ __launch_bounds__(256) void proj_gemm_kernel(
    const float* __restrict__ A, const float* __restrict__ wp,
    const float* __restrict__ bp, float* __restrict__ out)
{
    // phase1: At2[16][130] float2 (4160 floats) + Wt2[16][66] float2 (2112 floats)
    // phase2: outT[64][132] floats (8448 floats)
    __shared__ float2 smemv[4224];            // 8448 floats, 8B-aligned
    float* smem = (float*)smemv;
    float2* At2 = smemv;                       // 16*130 = 2080 float2
    float2* Wt2 = smemv + 16 * 130;            // 16*66  = 1056 float2
    const int mt = blockIdx.x;                 // 256 tiles of 128 rows
    const int nt = blockIdx.y;                 // 4 tiles of 64 cols
    const int row0 = mt * 128;
    const int j0 = nt * 64;
    const int tid = threadIdx.x, lane = tid & 31, wave = tid >> 5;
    const int m0 = wave * 16;
    const int laneM = lane & 15;
    const int laneH = lane >> 4;

    v8f acc[4] = {};
    for (int kc = 0; kc < CC; kc += 32) {
        for (int i = tid; i < 16 * 128; i += 256) {
            const int p = i & 15, rr = i >> 4;
            At2[p * 130 + rr] = *(const float2*)&A[(size_t)(row0 + rr) * CC + kc + 2 * p];
        }
        for (int i = tid; i < 16 * 64; i += 256) {
            const int p = i & 15, j = i >> 4;
            Wt2[p * 66 + j] = *(const float2*)&wp[(size_t)(j0 + j) * CC + kc + 2 * p];
        }
        __syncthreads();
#pragma unroll
        for (int kk = 0; kk < 32; kk += 4) {
            const int pp = (kk >> 1) + laneH;
            const v2f a = *(const v2f*)&At2[pp * 130 + m0 + laneM];
#pragma unroll
            for (int nb = 0; nb < 4; ++nb) {
                const v2f b = *(const v2f*)&Wt2[pp * 66 + nb * 16 + laneM];
                acc[nb] = wmma_f32(a, b, acc[nb]);
            }
        }
        __syncthreads();
    }
    // Stash D tiles transposed in LDS: outT[j][l], stride 132
    const int lbase = m0 + (laneH << 3);
#pragma unroll
    for (int nb = 0; nb < 4; ++nb) {
        const int j = nb * 16 + laneM;
#pragma unroll
        for (int r = 0; r < 8; ++r) smem[j * 132 + lbase + r] = acc[nb][r];
    }
    __syncthreads();
    const int n = row0 >> 12;
    const int lg0 = row0 & (LL - 1);
    for (int i = tid; i < 64 * 128; i += 256) {
        const int l = i & 127, j = i >> 7;
        out[((size_t)(n * CC + j0 + j)) * LL + lg0 + l] = smem[j * 132 + l] + bp[j0 + j];
    }
}

extern "C" void kernel_launch(void* const* d_in, const int* in_sizes, int n_in,
                              void* d_out, int out_size, void* d_ws, size_t ws_size,
                              hipStream_t stream) {
    const float* x      = (const float*)d_in[0];
    const float* w_qkv  = (const float*)d_in[1];
    const float* b_qkv  = (const float*)d_in[2];
    const float* w_proj = (const float*)d_in[3];
    const float* b_proj = (const float*)d_in[4];
    const float* w_dcv  = (const float*)d_in[5];
    float* out = (float*)d_out;

    float* qkv     = (float*)d_ws;                     // N*L*768  = 25,165,824 floats
    float* attn    = qkv + (size_t)NB * LL * TC;       // N*H*32*32 = 65,536 floats
    float* out_pre = attn + (size_t)NB * HH * DD * DD; // N*L*256  = 8,388,608 floats

    qkv_gemm_kernel<<<dim3(256, 12), 256, 0, stream>>>(x, w_qkv, b_qkv, qkv);
    norm_qk_kernel<<<2048, 256, 0, stream>>>(qkv);
    attn_kernel<<<64, 256, 0, stream>>>(qkv, attn);
    epilogue_kernel<<<4096, 256, 0, stream>>>(qkv, attn, w_dcv, out_pre);
    proj_gemm_kernel<<<dim3(256, 4), 256, 0, stream>>>(out_pre, w_proj, b_proj, out);
}